// BaselineModel_47682726921062
// MI455X (gfx1250) — compile-verified
//
#include <hip/hip_runtime.h>
#include <math.h>

// Problem dims (match reference)
#define BATCH 64
#define SEQ   512
#define DIM   512
#define BT    (BATCH * SEQ)        // 32768 rows of the front GEMM
#define BH    (BATCH * DIM)        // 32768
#define HT    (DIM * SEQ)

#define LDA   (DIM + 4)            // LDS pad: 516 -> bank-conflict-free frag reads

typedef __attribute__((ext_vector_type(2))) float v2f;
typedef __attribute__((ext_vector_type(8))) float v8f;

// ---------------------------------------------------------------------------
// f32 WMMA helper: D = A(16x4) * B(4x16) + C, CDNA5 V_WMMA_F32_16X16X4_F32
// ---------------------------------------------------------------------------
__device__ __forceinline__ v8f wmma_f32_4(v2f a, v2f b, v8f c) {
  return __builtin_amdgcn_wmma_f32_16x16x4_f32(
      /*neg_a=*/false, a, /*neg_b=*/false, b,
      /*c_mod=*/(short)0, c, /*reuse_a=*/false, /*reuse_b=*/false);
}

// ---------------------------------------------------------------------------
// CDNA5 async global->LDS copy (ASYNCcnt path, no VGPR data round-trip).
// Low 32 bits of a flat shared pointer == wave-relative LDS byte address.
// ---------------------------------------------------------------------------
__device__ __forceinline__ void async_copy_b128(float* lds_dst, const float* gsrc) {
  unsigned ldsa = (unsigned)(uintptr_t)lds_dst;
  asm volatile("global_load_async_to_lds_b128 %0, %1, off"
               :: "v"(ldsa), "v"(gsrc)
               : "memory");
}
__device__ __forceinline__ void async_wait0() {
  asm volatile("s_wait_asynccnt 0" ::: "memory");
}

// ---------------------------------------------------------------------------
// Grid-wide generation barrier (bar[0]=arrive count, bar[1]=generation)
// ---------------------------------------------------------------------------
__device__ __forceinline__ void grid_barrier(int* bar, int nblocks) {
  __syncthreads();
  __threadfence();
  if (threadIdx.x == 0) {
    int gen = __hip_atomic_load(bar + 1, __ATOMIC_RELAXED, __HIP_MEMORY_SCOPE_AGENT);
    int prev = __hip_atomic_fetch_add(bar, 1, __ATOMIC_ACQ_REL, __HIP_MEMORY_SCOPE_AGENT);
    if (prev == nblocks - 1) {
      __hip_atomic_store(bar, 0, __ATOMIC_RELAXED, __HIP_MEMORY_SCOPE_AGENT);
      __hip_atomic_fetch_add(bar + 1, 1, __ATOMIC_RELEASE, __HIP_MEMORY_SCOPE_AGENT);
    } else {
      while (__hip_atomic_load(bar + 1, __ATOMIC_ACQUIRE, __HIP_MEMORY_SCOPE_AGENT) == gen) {
        __builtin_amdgcn_s_sleep(8);
      }
    }
  }
  __syncthreads();
}

// ---------------------------------------------------------------------------
// Kernel 1: fused embedding gather + input projection GEMM (+ bias)
//   xp[(b*SEQ + t)*DIM + h] = sum_k emb[x[b,t],k]*W_ih[h,k] + b_ih[h] + b_hh[h]
// grid = (8 n-groups, 2048 m-tiles), block = 128 (4 waves, each a 16x16 tile)
// ---------------------------------------------------------------------------
__global__ __launch_bounds__(128) void embed_proj_wmma(
    const int* __restrict__ x, const float* __restrict__ emb,
    const float* __restrict__ Wih, const float* __restrict__ bih,
    const float* __restrict__ bhh, float* __restrict__ xp) {
  __shared__ float As[16 * LDA];
  __shared__ int rows[16];

  const int tid   = threadIdx.x;
  const int wave  = tid >> 5;
  const int lane  = tid & 31;
  const int mtile = blockIdx.y;                // 0..2047
  const int n0    = blockIdx.x * 64 + wave * 16;
  const int col   = n0 + (lane & 15);          // output column h
  const int khalf = (lane >> 4) << 1;          // 0 or 2 (K sub-offset per ISA layout)
  const int row   = lane & 15;

  if (tid < 16) rows[tid] = x[mtile * 16 + tid];   // x flat index == b*SEQ+t == gemm row
  __syncthreads();

  // Stage 16 gathered embedding rows (16 x 512 f32) into LDS via async copies
#pragma unroll
  for (int it = 0; it < 16; ++it) {
    const int idx = tid + it * 128;              // 2048 float4 chunks
    const int r   = idx >> 7;
    const int c4  = idx & 127;
    async_copy_b128(&As[r * LDA + c4 * 4], &emb[(size_t)rows[r] * DIM + c4 * 4]);
  }
  async_wait0();
  __syncthreads();

  const float2* bp = (const float2*)(Wih + (size_t)col * DIM);
  v8f c = {};
  for (int ko = 0; ko < 8; ++ko) {
    __builtin_prefetch((const void*)(bp + ko * 32 + 64), 0, 3);
#pragma unroll
    for (int ki = 0; ki < 16; ++ki) {
      const int kk = ko * 16 + ki;
      const int kb = kk * 4 + khalf;
      v2f a; a.x = As[row * LDA + kb]; a.y = As[row * LDA + kb + 1];
      const float2 bf = bp[kk * 2 + (khalf >> 1)];
      v2f b; b.x = bf.x; b.y = bf.y;
      c = wmma_f32_4(a, b, c);
    }
  }

  const float bias = bih[col] + bhh[col];
#pragma unroll
  for (int r8 = 0; r8 < 8; ++r8) {
    const int m = r8 + ((lane >> 4) << 3);
    const size_t grow = (size_t)mtile * 16 + m;      // = b*SEQ + t
    xp[grow * DIM + col] = c[r8] + bias;
  }
}

// ---------------------------------------------------------------------------
// Kernel 2: persistent RNN. 32 blocks x 128 threads; each wave owns one 16x16
// tile of h_t and pins its 16x512 W_hh slice in 256 VGPRs for all 512 steps.
// The K loop is FULLY unrolled so breg[] stays in registers (no scratch).
//   hid[t][b][h] = tanh(xp[b][t][h] + sum_j hid[t-1][b][j]*W_hh[h][j])
// ---------------------------------------------------------------------------
__global__ __launch_bounds__(128, 1) void rnn_persistent(
    const float* __restrict__ xp, const float* __restrict__ Whh,
    float* __restrict__ hid, int* __restrict__ bar) {
  __shared__ float As[16 * LDA];

  const int tid   = threadIdx.x;
  const int wave  = tid >> 5;
  const int lane  = tid & 31;
  const int mtile = blockIdx.x >> 3;               // 0..3 (batch tile)
  const int ng    = blockIdx.x & 7;                // 0..7
  const int n0    = ng * 64 + wave * 16;
  const int col   = n0 + (lane & 15);
  const int khalf = (lane >> 4) << 1;
  const int row   = lane & 15;
  const int nblocks = gridDim.x;

  // Pin this wave's B-operand (W_hh[col][:]) in registers: 128 x float2.
  float2 breg[128];
#pragma unroll
  for (int kk = 0; kk < 128; ++kk)
    breg[kk] = *(const float2*)&Whh[(size_t)col * DIM + kk * 4 + khalf];

  for (int t = 0; t < SEQ; ++t) {
    v8f c = {};
    if (t > 0) {
      // Stage h_{t-1} rows for this batch tile (16 x 512 f32) via async copies
      const float* src = hid + (size_t)(t - 1) * BH + (size_t)mtile * 16 * DIM;
#pragma unroll
      for (int it = 0; it < 16; ++it) {
        const int idx = tid + it * 128;
        const int r   = idx >> 7;
        const int c4  = idx & 127;
        async_copy_b128(&As[r * LDA + c4 * 4], &src[r * DIM + c4 * 4]);
      }
      async_wait0();
      __syncthreads();
#pragma unroll
      for (int kk = 0; kk < 128; ++kk) {       // fully unrolled: breg stays in VGPRs
        const int kb = kk * 4 + khalf;
        v2f a; a.x = As[row * LDA + kb]; a.y = As[row * LDA + kb + 1];
        v2f b; b.x = breg[kk].x; b.y = breg[kk].y;
        c = wmma_f32_4(a, b, c);
      }
    }
    // Epilogue: h = tanh(xp + h_prev @ W_hh^T); write hiddens[t]
#pragma unroll
    for (int r8 = 0; r8 < 8; ++r8) {
      const int m  = r8 + ((lane >> 4) << 3);
      const int gb = mtile * 16 + m;                           // batch index
      const float xpv = xp[((size_t)gb * SEQ + t) * DIM + col];
      hid[(size_t)t * BH + (size_t)gb * DIM + col] = tanhf(xpv + c[r8]);
    }
    if (t != SEQ - 1) grid_barrier(bar, nblocks);
  }
}

// ---------------------------------------------------------------------------
// Kernel 3: softmax over H + [T,B,H] -> [B,H,T] transpose.
// block = 512 (16 waves); wave w handles t = blockIdx.y*16 + w.
// ---------------------------------------------------------------------------
__global__ __launch_bounds__(512) void softmax_transpose(
    const float* __restrict__ hid, float* __restrict__ out) {
  __shared__ float Ts[DIM * 17];

  const int tid  = threadIdx.x;
  const int wave = tid >> 5;
  const int lane = tid & 31;
  const int b    = blockIdx.x;           // 0..63
  const int t0   = blockIdx.y * 16;      // 0,16,...,496
  const int t    = t0 + wave;

  const float* src = hid + (size_t)t * BH + (size_t)b * DIM;
  float v[16];
  float m = -3.402823466e+38f;
#pragma unroll
  for (int i = 0; i < 16; ++i) {
    v[i] = src[lane + 32 * i];
    m = fmaxf(m, v[i]);
  }
#pragma unroll
  for (int off = 16; off >= 1; off >>= 1) m = fmaxf(m, __shfl_xor(m, off, 32));
  float s = 0.f;
#pragma unroll
  for (int i = 0; i < 16; ++i) { v[i] = __expf(v[i] - m); s += v[i]; }
#pragma unroll
  for (int off = 16; off >= 1; off >>= 1) s += __shfl_xor(s, off, 32);
  const float inv = 1.0f / s;
#pragma unroll
  for (int i = 0; i < 16; ++i) Ts[(lane + 32 * i) * 17 + wave] = v[i] * inv;
  __syncthreads();

  // thread j owns h=j: write out[b][h][t0..t0+15] as 4 contiguous float4
  const int h = tid;
  float* dst = out + (size_t)b * HT + (size_t)h * SEQ + t0;
#pragma unroll
  for (int q = 0; q < 4; ++q) {
    float4 o;
    o.x = Ts[h * 17 + q * 4 + 0];
    o.y = Ts[h * 17 + q * 4 + 1];
    o.z = Ts[h * 17 + q * 4 + 2];
    o.w = Ts[h * 17 + q * 4 + 3];
    *(float4*)(dst + q * 4) = o;
  }
}

// ---------------------------------------------------------------------------
extern "C" void kernel_launch(void* const* d_in, const int* in_sizes, int n_in,
                              void* d_out, int out_size, void* d_ws, size_t ws_size,
                              hipStream_t stream) {
  (void)in_sizes; (void)n_in; (void)out_size; (void)ws_size;
  const int*   x    = (const int*)d_in[0];
  const float* emb  = (const float*)d_in[1];
  const float* Wih  = (const float*)d_in[2];
  const float* Whh  = (const float*)d_in[3];
  const float* bih  = (const float*)d_in[4];
  const float* bhh  = (const float*)d_in[5];
  float* out = (float*)d_out;

  // xp lives in d_out (64MB, fully overwritten later by softmax — stream-ordered).
  float* xp  = (float*)d_out;
  float* hid = (float*)d_ws;                         // 64MB hiddens [T][B][H]
  int*   bar = (int*)((char*)d_ws + (size_t)BT * DIM * sizeof(float));

  hipMemsetAsync(bar, 0, 2 * sizeof(int), stream);   // reset barrier (capture-legal)

  embed_proj_wmma<<<dim3(8, BT / 16), 128, 0, stream>>>(x, emb, Wih, bih, bhh, xp);
  rnn_persistent<<<dim3(32), 128, 0, stream>>>(xp, Whh, hid, bar);
  softmax_transpose<<<dim3(BATCH, SEQ / 16), 512, 0, stream>>>(hid, out);
}